// DoublyStochasticLayer_18537078849765
// MI455X (gfx1250) — compile-verified
//
#include <hip/hip_runtime.h>
#include <hip/hip_bf16.h>

typedef __attribute__((ext_vector_type(2))) float v2f;
typedef __attribute__((ext_vector_type(8))) float v8f;

#define ITERS   20
#define EPS     1e-8f
#define NROW    256
#define NCOL    256
#define STRIDE  260          // padded LDS row stride (floats) -> conflict-free ds_load_b64 per row
#define TPB     512          // 16 wave32s per workgroup

// One workgroup per 256x256 matrix. Matrix lives in LDS for all 20 Sinkhorn
// iterations; row/col sums computed with V_WMMA_F32_16X16X4_F32 against an
// all-ones operand (sum is invariant to K-permutation of the operand layout).
extern "C" __global__ __launch_bounds__(TPB)
void sinkhorn_doubly_stochastic(const float* __restrict__ in, float* __restrict__ out) {
    extern __shared__ float smem[];
    float* Mat    = smem;                    // NROW * STRIDE floats (padded matrix)
    float* rowinv = smem + NROW * STRIDE;    // 256 floats: 1/(rowsum+eps)
    float* colinv = rowinv + NROW;           // 256 floats: 1/(colsum+eps)
    float* redbuf = rowinv;                  // 512-float scratch reused for max reduce

    const int tid  = threadIdx.x;
    const int lane = tid & 31;
    const int wave = tid >> 5;               // 0..15
    const int half = lane >> 4;              // 0 or 1 (half-wave)

    const size_t mbase = (size_t)blockIdx.x * (NROW * NCOL);
    const float4* gin  = (const float4*)(in  + mbase);
    float4*       gout = (float4*)(out + mbase);

    // ---------- Phase 0: global -> LDS (padded), track local max ----------
    float lmax = -3.402823466e38f;
    #pragma unroll 4
    for (int i = 0; i < 32; ++i) {
        int f = tid + i * TPB;               // float4 index 0..16383
        float4 v = gin[f];
        lmax = fmaxf(lmax, fmaxf(fmaxf(v.x, v.y), fmaxf(v.z, v.w)));
        int e = f << 2;                      // element index
        int r = e >> 8;                      // /256
        int c = e & 255;
        *(float4*)(Mat + r * STRIDE + c) = v;
    }
    redbuf[tid] = lmax;
    __syncthreads();
    for (int s = TPB / 2; s > 0; s >>= 1) {
        if (tid < s) redbuf[tid] = fmaxf(redbuf[tid], redbuf[tid + s]);
        __syncthreads();
    }
    const float gmax = redbuf[0];            // read-before-exp; barrier after exp
                                             // protects redbuf reuse as rowinv
    // ---------- exp(x - max), in place ----------
    const int myrow = tid >> 1;              // 2 threads per row
    const int mycb  = (tid & 1) << 7;        // column base: 0 or 128
    {
        float4* p = (float4*)(Mat + myrow * STRIDE + mycb);
        #pragma unroll 4
        for (int j = 0; j < 32; ++j) {
            float4 v = p[j];
            v.x = __expf(v.x - gmax);
            v.y = __expf(v.y - gmax);
            v.z = __expf(v.z - gmax);
            v.w = __expf(v.w - gmax);
            p[j] = v;
        }
    }
    __syncthreads();

    // A-operand pointer for row sums: lane -> row (M = lane%16), half-wave -> K offset
    const int   mrow = (wave << 4) + (lane & 15);
    const float* arp = Mat + mrow * STRIDE + (half << 1);
    // B-operand addressing for col sums: lane -> column (N = lane%16), half-wave -> K rows
    const int   ncol = (wave << 4) + (lane & 15);
    const int   kr   = half << 1;            // 0 or 2 within each 4-row chunk
    const v2f one2 = {1.0f, 1.0f};

    for (int it = 0; it < ITERS; ++it) {
        // ---- row sums: D = A(16x4 chunk) x ones(4x16), 64 K-chunks ----
        v8f acc = {};
        #pragma unroll 4
        for (int c = 0; c < 64; ++c) {
            v2f a = *(const v2f*)(arp + (c << 2));   // 2 contiguous floats (ds_load_b64)
            acc = __builtin_amdgcn_wmma_f32_16x16x4_f32(
                      false, a, false, one2, (short)0, acc, false, false);
        }
        if ((lane & 15) == 0) {                      // lanes 0 and 16 hold M=g / M=8+g
            int rb = (wave << 4) + (half << 3);
            #pragma unroll
            for (int g = 0; g < 8; ++g)
                rowinv[rb + g] = 1.0f / (acc[g] + EPS);
        }
        __syncthreads();

        // ---- scale rows ----
        {
            float s = rowinv[myrow];
            float4* p = (float4*)(Mat + myrow * STRIDE + mycb);
            #pragma unroll 8
            for (int j = 0; j < 32; ++j) {
                float4 v = p[j];
                v.x *= s; v.y *= s; v.z *= s; v.w *= s;
                p[j] = v;
            }
        }
        __syncthreads();

        // ---- col sums: D = ones(16x4) x B(4x16 chunk), 64 K-chunks ----
        acc = (v8f){};
        const float* cbp = Mat + kr * STRIDE + ncol;
        #pragma unroll 4
        for (int c = 0; c < 64; ++c) {
            v2f b;
            b.x = cbp[(c << 2) * STRIDE];            // row kr   of chunk
            b.y = cbp[(c << 2) * STRIDE + STRIDE];   // row kr+1 of chunk
            acc = __builtin_amdgcn_wmma_f32_16x16x4_f32(
                      false, one2, false, b, (short)0, acc, false, false);
        }
        if (lane < 16)                               // VGPR0, M=0 row: colsum[N=lane]
            colinv[ncol] = 1.0f / (acc[0] + EPS);
        __syncthreads();

        // ---- scale cols ----
        {
            float4* p        = (float4*)(Mat + myrow * STRIDE + mycb);
            const float4* ci = (const float4*)(colinv + mycb);
            #pragma unroll 8
            for (int j = 0; j < 32; ++j) {
                float4 v = p[j];
                float4 s = ci[j];
                v.x *= s.x; v.y *= s.y; v.z *= s.z; v.w *= s.w;
                p[j] = v;
            }
        }
        __syncthreads();
    }

    // ---------- writeback: LDS -> global ----------
    #pragma unroll 4
    for (int i = 0; i < 32; ++i) {
        int f = tid + i * TPB;
        int e = f << 2;
        int r = e >> 8;
        int c = e & 255;
        gout[f] = *(const float4*)(Mat + r * STRIDE + c);
    }
}

extern "C" void kernel_launch(void* const* d_in, const int* in_sizes, int n_in,
                              void* d_out, int out_size, void* d_ws, size_t ws_size,
                              hipStream_t stream) {
    (void)n_in; (void)d_ws; (void)ws_size; (void)out_size;
    const float* in = (const float*)d_in[0];
    float* out      = (float*)d_out;
    const int nmat  = in_sizes[0] / (NROW * NCOL);   // 64*12 = 768 matrices

    const size_t shmem = (size_t)(NROW * STRIDE + 2 * NROW) * sizeof(float); // ~262 KB

    // Allow >64KB dynamic LDS (CDNA5 WGP has 320KB). Deterministic, idempotent.
    (void)hipFuncSetAttribute((const void*)sinkhorn_doubly_stochastic,
                              hipFuncAttributeMaxDynamicSharedMemorySize, (int)shmem);

    sinkhorn_doubly_stochastic<<<nmat, TPB, shmem, stream>>>(in, out);
}